// SpectralSonar_39986145526416
// MI455X (gfx1250) — compile-verified
//
#include <hip/hip_runtime.h>

// SpectralSonar on MI455X (gfx1250, wave32).
// Strategy: per-wave, 16 parameter rows; ghost values computed with hardware
// transcendentals (v_sin/v_cos/v_exp/v_log); the three row reductions
// (sum g, sum g^2, sum g*t_std) are offloaded to the matrix pipe via
// V_WMMA_F32_16X16X4_F32 (A = 16x4 ghost tile, B cols = [ones | t_std | 0...]).
// Argmin via packed u64 atomicMin (first-occurrence tie-break preserved).

typedef __attribute__((ext_vector_type(2))) float v2f;
typedef __attribute__((ext_vector_type(8))) float v8f;

#define LREF 2048
#define EPS_F 1e-6f
static constexpr float PI_F     = 3.14159265358979323846f;
static constexpr float LOG2_10F = 3.3219280948873623f;

// exp2 on the trans pipe: v_exp_f32
__device__ __forceinline__ float fast_exp2(float x) {
    return __builtin_amdgcn_exp2f(x);
}

__device__ __forceinline__ float ghost_val(float phi, float m, float n2, float n3,
                                           float negInvN1) {
    float ang = 0.25f * m * phi;
    float s, c;
    __sincosf(ang, &s, &c);
    c = fabsf(c);
    s = fabsf(s);
    // |cos|^n2 + |sin|^n3 via exp2/log2 (log2(0) -> -inf -> exp2 -> 0, correct)
    float p2 = fast_exp2(n2 * __log2f(c));
    float p3 = fast_exp2(n3 * __log2f(s));
    float R  = fast_exp2(negInvN1 * __log2f(p2 + p3));
    return R * __sinf(phi);
}

__device__ __forceinline__ unsigned long long shflxor_u64(unsigned long long v, int mask) {
    unsigned int lo = (unsigned int)(v & 0xffffffffULL);
    unsigned int hi = (unsigned int)(v >> 32);
    lo = __shfl_xor(lo, mask, 32);
    hi = __shfl_xor(hi, mask, 32);
    return ((unsigned long long)hi << 32) | (unsigned long long)lo;
}

// One wave sweeps 16 parameter rows over all L samples, accumulating with WMMA.
// lane&15 selects the row; lanes 16..31 mirror rows for K=2,3 of the A tile.
__device__ __forceinline__ void sweep16(
    int lane, int rowIdx, int numRows,
    float m, float n1, float n2, float n3,
    float k0, float k1,
    const float* __restrict__ tstd, float St, float Stt,
    unsigned long long* slot)
{
    const float negInvN1 = -1.0f / n1;
    const int lhalf = (lane >> 4) << 1;   // 0 for lanes 0-15 (K=0,1), 2 for 16-31 (K=2,3)
    const int col   = lane & 15;          // B-matrix column owned by this lane

    v8f acc1 = {0.f,0.f,0.f,0.f,0.f,0.f,0.f,0.f};  // col0 = sum g,   col1 = sum g*t
    v8f acc2 = {0.f,0.f,0.f,0.f,0.f,0.f,0.f,0.f};  // col0 = sum g^2

    #pragma unroll 4
    for (int base = 0; base < LREF; base += 4) {
        int   l0   = base + lhalf;
        float phi0 = fmaf((float)l0, k1, k0);
        float phi1 = phi0 + k1;
        float g0 = ghost_val(phi0, m, n2, n3, negInvN1);
        float g1 = ghost_val(phi1, m, n2, n3, negInvN1);
        float tv0 = tstd[l0];
        float tv1 = tstd[l0 + 1];
        // B: row K in {l0,l0+1}; col0 = 1, col1 = t_std[K], rest 0 (cndmask, no branches)
        float b0 = (col == 0) ? 1.0f : ((col == 1) ? tv0 : 0.0f);
        float b1 = (col == 0) ? 1.0f : ((col == 1) ? tv1 : 0.0f);
        v2f A  = { g0, g1 };
        v2f A2 = { g0 * g0, g1 * g1 };
        v2f B  = { b0, b1 };
        acc1 = __builtin_amdgcn_wmma_f32_16x16x4_f32(false, A,  false, B, (short)0, acc1, false, false);
        acc2 = __builtin_amdgcn_wmma_f32_16x16x4_f32(false, A2, false, B, (short)0, acc2, false, false);
    }

    // Extract per-row sums from the 16x16 accumulators:
    // D vgpr v: lanes 0-15 hold row v (col = lane), lanes 16-31 hold row v+8.
    float Sg = 0.f, Sgt = 0.f, Sgg = 0.f;
    #pragma unroll
    for (int v = 0; v < 8; ++v) {
        float a = acc1[v], q = acc2[v];
        float sgL  = __shfl(a, 0, 32),  sgtL = __shfl(a, 1, 32),  sggL = __shfl(q, 0, 32);
        float sgH  = __shfl(a, 16, 32), sgtH = __shfl(a, 17, 32), sggH = __shfl(q, 16, 32);
        if (lane == v)     { Sg = sgL; Sgt = sgtL; Sgg = sggL; }
        if (lane == v + 8) { Sg = sgH; Sgt = sgtH; Sgg = sggH; }
    }

    // MSE of standardized ghost vs standardized target, expanded in sums:
    const float invL   = 1.0f / (float)LREF;
    const float invLm1 = 1.0f / (float)(LREF - 1);
    float gbar  = Sg * invL;
    float ss    = Sgg - Sg * gbar;                 // sum (g - gbar)^2
    float sd    = sqrtf(fmaxf(ss * invLm1, 0.0f)); // ddof=1 std
    float denom = sd + EPS_F;
    float term1 = ss / (denom * denom);
    float cross = (Sgt - gbar * St) / denom;
    float mse   = fmaxf((term1 - 2.0f * cross + Stt) * invL, 0.0f);

    unsigned long long packed =
        ((unsigned long long)__float_as_uint(mse) << 32) | (unsigned long long)(unsigned int)rowIdx;
    if (lane >= 16 || rowIdx >= numRows) packed = ~0ULL;

    #pragma unroll
    for (int off = 16; off > 0; off >>= 1) {
        unsigned long long o = shflxor_u64(packed, off);
        packed = (o < packed) ? o : packed;
    }
    if (lane == 0) atomicMin(slot, packed);
}

// ---------------- prep: standardize target, constants, reset argmin slots ---
__global__ void prep_kernel(const float* __restrict__ x, float* __restrict__ tstd,
                            float* __restrict__ stats, unsigned long long* __restrict__ slots) {
    __shared__ float red[256];
    int tid = threadIdx.x;
    float s = 0.f, s2 = 0.f;
    for (int i = tid; i < LREF; i += 256) { float v = x[i]; s += v; s2 = fmaf(v, v, s2); }
    red[tid] = s; __syncthreads();
    for (int o = 128; o > 0; o >>= 1) { if (tid < o) red[tid] += red[tid + o]; __syncthreads(); }
    float Sx = red[0]; __syncthreads();
    red[tid] = s2; __syncthreads();
    for (int o = 128; o > 0; o >>= 1) { if (tid < o) red[tid] += red[tid + o]; __syncthreads(); }
    float Sxx = red[0]; __syncthreads();

    float mean  = Sx / (float)LREF;
    float var   = (Sxx - Sx * mean) / (float)(LREF - 1);
    float denom = sqrtf(fmaxf(var, 0.0f)) + EPS_F;

    float st = 0.f, stt = 0.f;
    for (int i = tid; i < LREF; i += 256) {
        float v = (x[i] - mean) / denom;
        tstd[i] = v;
        st += v; stt = fmaf(v, v, stt);
    }
    red[tid] = st; __syncthreads();
    for (int o = 128; o > 0; o >>= 1) { if (tid < o) red[tid] += red[tid + o]; __syncthreads(); }
    float StT = red[0]; __syncthreads();
    red[tid] = stt; __syncthreads();
    for (int o = 128; o > 0; o >>= 1) { if (tid < o) red[tid] += red[tid + o]; __syncthreads(); }
    float SttT = red[0];

    if (tid == 0) {
        stats[0] = StT;
        stats[1] = SttT;
        slots[0] = ~0ULL;
        slots[1] = ~0ULL;
    }
}

// ---------------- coarse pass: 80,000 rows = 5000 waves x 16 rows -----------
__global__ void coarse_kernel(const int* __restrict__ freqp, const float* __restrict__ tstd,
                              const float* __restrict__ stats, unsigned long long* __restrict__ slots) {
    int lane = threadIdx.x & 31;
    int wave = blockIdx.x * (blockDim.x >> 5) + (threadIdx.x >> 5);
    int r  = wave * 16 + (lane & 15);
    int rc = (r < 80000) ? r : 79999;
    int mi = rc / 8000;
    int i1 = (rc / 400) % 20;
    int i2 = (rc / 20) % 20;
    int i3 = rc % 20;
    float m  = (float)(mi + 1);
    float n1 = fast_exp2(LOG2_10F * (-1.0f + 2.0f * (float)i1 / 19.0f));
    float n2 = fast_exp2(LOG2_10F * (-1.0f + 2.0f * (float)i2 / 19.0f));
    float n3 = fast_exp2(LOG2_10F * (-1.0f + 2.0f * (float)i3 / 19.0f));
    float F  = (float)freqp[0];
    float k1 = 4.0f * PI_F * F / (float)(LREF - 1);
    float k0 = -2.0f * PI_F * F;
    sweep16(lane, r, 80000, m, n1, n2, n3, k0, k1, tstd, stats[0], stats[1], &slots[0]);
}

// ---------------- fine pass: 27,000 rows around the coarse winner -----------
__global__ void fine_kernel(const int* __restrict__ freqp, const float* __restrict__ tstd,
                            const float* __restrict__ stats, unsigned long long* __restrict__ slots) {
    int lane = threadIdx.x & 31;
    int wave = blockIdx.x * (blockDim.x >> 5) + (threadIdx.x >> 5);

    unsigned int cr = (unsigned int)(slots[0] & 0xffffffffULL);
    int mi = (int)(cr / 8000u);
    int i1 = (int)((cr / 400u) % 20u);
    int i2 = (int)((cr / 20u) % 20u);
    int i3 = (int)(cr % 20u);
    float c_m  = (float)(mi + 1);
    float c_n1 = fast_exp2(LOG2_10F * (-1.0f + 2.0f * (float)i1 / 19.0f));
    float c_n2 = fast_exp2(LOG2_10F * (-1.0f + 2.0f * (float)i2 / 19.0f));
    float c_n3 = fast_exp2(LOG2_10F * (-1.0f + 2.0f * (float)i3 / 19.0f));

    int r  = wave * 16 + (lane & 15);
    int rc = (r < 27000) ? r : 26999;
    int f1 = rc / 900;
    int f2 = (rc / 30) % 30;
    int f3 = rc % 30;
    float n1 = c_n1 * (0.8f + 0.4f * (float)f1 / 29.0f);
    float n2 = c_n2 * (0.8f + 0.4f * (float)f2 / 29.0f);
    float n3 = c_n3 * (0.8f + 0.4f * (float)f3 / 29.0f);
    float F  = (float)freqp[0];
    float k1 = 4.0f * PI_F * F / (float)(LREF - 1);
    float k0 = -2.0f * PI_F * F;
    sweep16(lane, r, 27000, c_m, n1, n2, n3, k0, k1, tstd, stats[0], stats[1], &slots[1]);
}

// ---------------- finalize: write [c_m, n1_f, n2_f, n3_f] -------------------
__global__ void finalize_kernel(const unsigned long long* __restrict__ slots,
                                float* __restrict__ out) {
    unsigned int cr = (unsigned int)(slots[0] & 0xffffffffULL);
    unsigned int fr = (unsigned int)(slots[1] & 0xffffffffULL);
    int mi = (int)(cr / 8000u);
    int i1 = (int)((cr / 400u) % 20u);
    int i2 = (int)((cr / 20u) % 20u);
    int i3 = (int)(cr % 20u);
    float c_n1 = fast_exp2(LOG2_10F * (-1.0f + 2.0f * (float)i1 / 19.0f));
    float c_n2 = fast_exp2(LOG2_10F * (-1.0f + 2.0f * (float)i2 / 19.0f));
    float c_n3 = fast_exp2(LOG2_10F * (-1.0f + 2.0f * (float)i3 / 19.0f));
    int f1 = (int)(fr / 900u);
    int f2 = (int)((fr / 30u) % 30u);
    int f3 = (int)(fr % 30u);
    out[0] = (float)(mi + 1);
    out[1] = c_n1 * (0.8f + 0.4f * (float)f1 / 29.0f);
    out[2] = c_n2 * (0.8f + 0.4f * (float)f2 / 29.0f);
    out[3] = c_n3 * (0.8f + 0.4f * (float)f3 / 29.0f);
}

extern "C" void kernel_launch(void* const* d_in, const int* in_sizes, int n_in,
                              void* d_out, int out_size, void* d_ws, size_t ws_size,
                              hipStream_t stream) {
    (void)in_sizes; (void)n_in; (void)out_size; (void)ws_size;
    const float* x     = (const float*)d_in[0];
    const int*   freqp = (const int*)d_in[1];

    // Workspace layout: [0..2047] t_std | [2048..2049] St,Stt | byte 8200: 2x u64 argmin
    float* wsf   = (float*)d_ws;
    float* tstd  = wsf;
    float* stats = wsf + LREF;
    unsigned long long* slots = (unsigned long long*)((char*)d_ws + 8200);

    prep_kernel<<<1, 256, 0, stream>>>(x, tstd, stats, slots);
    // coarse: 80000 rows / 16 per wave = 5000 waves; 4 waves per 128-thread block
    coarse_kernel<<<1250, 128, 0, stream>>>(freqp, tstd, stats, slots);
    // fine: ceil(27000/16) = 1688 waves = 422 blocks x 4 waves
    fine_kernel<<<422, 128, 0, stream>>>(freqp, tstd, stats, slots);
    finalize_kernel<<<1, 1, 0, stream>>>(slots, (float*)d_out);
}